// mLSTM_43370579755367
// MI455X (gfx1250) — compile-verified
//
#include <hip/hip_runtime.h>
#include <hip/hip_bf16.h>
#include <math.h>

typedef __attribute__((ext_vector_type(2))) float v2f;
typedef __attribute__((ext_vector_type(4))) float v4f;
typedef __attribute__((ext_vector_type(8))) float v8f;

#define Bsz 64
#define Isz 512
#define Msz 1024

// ---------------------------------------------------------------------------
// Kernel 1: fused 4-way GEMM  out[n_mat][64,1024] = x[64,512] @ W[n_mat]^T + b
// using V_WMMA_F32_16X16X4_F32 (full fp32). One wave computes one 16x16 tile.
// Tiles: 4 (batch) x 64 (cols) x 4 (matrices) = 1024 tiles -> 128 blocks x 8 waves.
// Epilogue fuses: +bias, k-scale (1/sqrt(1024)), sigmoid for o-gate.
// ---------------------------------------------------------------------------
__global__ __launch_bounds__(256) void mlstm_gemm_wmma(
    const float* __restrict__ x,
    const float* __restrict__ Wq, const float* __restrict__ bq,
    const float* __restrict__ Wk, const float* __restrict__ bk,
    const float* __restrict__ Wv, const float* __restrict__ bv,
    const float* __restrict__ Wo, const float* __restrict__ bo,
    float* __restrict__ qt, float* __restrict__ kt,
    float* __restrict__ vt, float* __restrict__ ot) {
  const int t = threadIdx.x;
  const int wave = t >> 5;
  const int lane = t & 31;
  const int tileId = blockIdx.x * 8 + wave;   // 0..1023
  const int mt = tileId & 3;                  // batch tile (4 x 16 = 64)
  const int nt = (tileId >> 2) & 63;          // col tile   (64 x 16 = 1024)
  const int mi = tileId >> 8;                 // which matrix 0..3

  const float* W; const float* bias; float* outp;
  if (mi == 0)      { W = Wq; bias = bq; outp = qt; }
  else if (mi == 1) { W = Wk; bias = bk; outp = kt; }
  else if (mi == 2) { W = Wv; bias = bv; outp = vt; }
  else              { W = Wo; bias = bo; outp = ot; }

  const int half = lane >> 4;     // 0: K=0,1  1: K=2,3 (per 16x4 f32 A layout)
  const int l16  = lane & 15;

  // A: x row (batch m = mt*16 + l16), two consecutive K per lane
  const float* arow = x + (size_t)(mt * 16 + l16) * Isz + 2 * half;
  // B[k][n] = W[n][k]: W row (col n = nt*16 + l16), two consecutive K per lane
  const float* brow = W + (size_t)(nt * 16 + l16) * Isz + 2 * half;

  v8f acc = {};
  for (int k0 = 0; k0 < Isz; k0 += 4) {
    v2f a  = *(const v2f*)(arow + k0);
    v2f bb = *(const v2f*)(brow + k0);
    // 8 args: (neg_a, A, neg_b, B, c_mod, C, reuse_a, reuse_b)
    acc = __builtin_amdgcn_wmma_f32_16x16x4_f32(
        false, a, false, bb, (short)0, acc, false, false);
  }

  const int n = nt * 16 + l16;
  const float bval = bias[n];
#pragma unroll
  for (int v = 0; v < 8; ++v) {
    const int m = mt * 16 + v + 8 * half;   // D row per 16x16 f32 C/D layout
    float val = acc[v] + bval;
    if (mi == 1)      val *= 0.03125f;                       // 1/sqrt(1024)
    else if (mi == 3) val = 1.0f / (1.0f + expf(-val));      // sigmoid(o)
    outp[(size_t)m * Msz + n] = val;
  }
}

// ---------------------------------------------------------------------------
// Kernel 2: scalar gates per batch: it = exp(x.wi + bi), ft = sigmoid(x.wf + bf)
// ---------------------------------------------------------------------------
__global__ __launch_bounds__(256) void mlstm_gates(
    const float* __restrict__ x,
    const float* __restrict__ wi, const float* __restrict__ bi,
    const float* __restrict__ wf, const float* __restrict__ bf,
    float* __restrict__ itp, float* __restrict__ ftp) {
  __shared__ float si[256];
  __shared__ float sf[256];
  const int b = blockIdx.x;
  const int t = threadIdx.x;
  const float* xr = x + (size_t)b * Isz;
  float pi = fmaf(xr[t], wi[t], xr[t + 256] * wi[t + 256]);
  float pf = fmaf(xr[t], wf[t], xr[t + 256] * wf[t + 256]);
  si[t] = pi; sf[t] = pf;
  __syncthreads();
  for (int s = 128; s > 0; s >>= 1) {
    if (t < s) { si[t] += si[t + s]; sf[t] += sf[t + s]; }
    __syncthreads();
  }
  if (t == 0) {
    itp[b] = expf(si[0] + bi[0]);
    ftp[b] = 1.0f / (1.0f + expf(-(sf[0] + bf[0])));
  }
}

// ---------------------------------------------------------------------------
// Kernel 3: n = ft*n_prev + it*kt  (writes n output) and inv = 1/max(|n.qt|,1)
// ---------------------------------------------------------------------------
__global__ __launch_bounds__(256) void mlstm_nstate(
    const float* __restrict__ n_prev, const float* __restrict__ kt,
    const float* __restrict__ qt,
    const float* __restrict__ itp, const float* __restrict__ ftp,
    float* __restrict__ n_out, float* __restrict__ invp) {
  __shared__ float sd[256];
  const int b = blockIdx.x;
  const int t = threadIdx.x;
  const float ft = ftp[b];
  const float it = itp[b];
  const float* np = n_prev + (size_t)b * Msz;
  const float* kp = kt + (size_t)b * Msz;
  const float* qp = qt + (size_t)b * Msz;
  float* no = n_out + (size_t)b * Msz;
  float dot = 0.0f;
  for (int j = t; j < Msz; j += 256) {
    float nv = fmaf(ft, np[j], it * kp[j]);
    no[j] = nv;
    dot = fmaf(nv, qp[j], dot);
  }
  sd[t] = dot;
  __syncthreads();
  for (int s = 128; s > 0; s >>= 1) {
    if (t < s) sd[t] += sd[t + s];
    __syncthreads();
  }
  if (t == 0) invp[b] = 1.0f / fmaxf(fabsf(sd[0]), 1.0f);
}

// ---------------------------------------------------------------------------
// Kernel 4: the bandwidth kernel. One wave per C row:
//   C[b,m,:] = ft*C_prev[b,m,:] + (it*vt[b,m]) * kt[b,:]
//   ht[b,m]  = ot[b,m] * (C[b,m,:].qt[b,:]) * inv[b]
// Non-temporal float4 stream for C (268 MB each way > 192 MB L2).
// 8 rows per block (8 waves), qt/kt rows staged in LDS.
// ---------------------------------------------------------------------------
__global__ __launch_bounds__(256) void mlstm_cstream(
    const float* __restrict__ C_prev,
    const float* __restrict__ qt, const float* __restrict__ kt,
    const float* __restrict__ vt, const float* __restrict__ ot,
    const float* __restrict__ itp, const float* __restrict__ ftp,
    const float* __restrict__ invp,
    float* __restrict__ C_out, float* __restrict__ ht) {
  __shared__ v4f q4[Msz / 4];
  __shared__ v4f k4[Msz / 4];
  const int b  = blockIdx.x >> 7;           // 128 blocks per batch
  const int m0 = (blockIdx.x & 127) << 3;   // 8 rows per block
  const int t  = threadIdx.x;

  q4[t] = *(const v4f*)(qt + (size_t)b * Msz + t * 4);
  k4[t] = *(const v4f*)(kt + (size_t)b * Msz + t * 4);
  __syncthreads();

  const float ftb  = ftp[b];
  const float itb  = itp[b];
  const float invb = invp[b];

  const int wave = t >> 5;
  const int lane = t & 31;
  const int m = m0 + wave;
  const float ikv = itb * vt[(size_t)b * Msz + m];

  const size_t rowoff = ((size_t)b * Msz + m) * Msz;
  const float* cp = C_prev + rowoff;
  float* co = C_out + rowoff;

  float acc = 0.0f;
#pragma unroll
  for (int iter = 0; iter < 8; ++iter) {
    const int j = iter * 128 + lane * 4;    // coalesced 512B per wave
    v4f c  = __builtin_nontemporal_load((const v4f*)(cp + j));
    v4f kv = k4[j >> 2];
    v4f qv = q4[j >> 2];
    v4f cn;
    cn.x = fmaf(ftb, c.x, ikv * kv.x);
    cn.y = fmaf(ftb, c.y, ikv * kv.y);
    cn.z = fmaf(ftb, c.z, ikv * kv.z);
    cn.w = fmaf(ftb, c.w, ikv * kv.w);
    __builtin_nontemporal_store(cn, (v4f*)(co + j));
    acc = fmaf(cn.x, qv.x, acc);
    acc = fmaf(cn.y, qv.y, acc);
    acc = fmaf(cn.z, qv.z, acc);
    acc = fmaf(cn.w, qv.w, acc);
  }

  // wave32 reduction
  for (int off = 16; off > 0; off >>= 1) acc += __shfl_xor(acc, off, 32);
  if (lane == 0) ht[(size_t)b * Msz + m] = ot[(size_t)b * Msz + m] * acc * invb;
}

// ---------------------------------------------------------------------------
extern "C" void kernel_launch(void* const* d_in, const int* in_sizes, int n_in,
                              void* d_out, int out_size, void* d_ws, size_t ws_size,
                              hipStream_t stream) {
  const float* x      = (const float*)d_in[0];
  const float* C_prev = (const float*)d_in[1];
  const float* n_prev = (const float*)d_in[2];
  const float* Wq = (const float*)d_in[3];  const float* bq = (const float*)d_in[4];
  const float* Wk = (const float*)d_in[5];  const float* bk = (const float*)d_in[6];
  const float* Wv = (const float*)d_in[7];  const float* bv = (const float*)d_in[8];
  const float* wi = (const float*)d_in[9];  const float* bi = (const float*)d_in[10];
  const float* wf = (const float*)d_in[11]; const float* bf = (const float*)d_in[12];
  const float* Wo = (const float*)d_in[13]; const float* bo = (const float*)d_in[14];

  // outputs: ht [B,H], C [B,M,M], n [B,M] concatenated
  float* out   = (float*)d_out;
  float* ht    = out;
  float* C     = out + (size_t)Bsz * Msz;
  float* n_out = out + (size_t)Bsz * Msz + (size_t)Bsz * Msz * Msz;

  // workspace layout (floats)
  float* ws  = (float*)d_ws;
  float* qt  = ws;
  float* kt  = ws + 1 * (size_t)Bsz * Msz;
  float* vt  = ws + 2 * (size_t)Bsz * Msz;
  float* ot  = ws + 3 * (size_t)Bsz * Msz;
  float* itp = ws + 4 * (size_t)Bsz * Msz;
  float* ftp = itp + Bsz;
  float* invp = ftp + Bsz;

  mlstm_gemm_wmma<<<128, 256, 0, stream>>>(x, Wq, bq, Wk, bk, Wv, bv, Wo, bo,
                                           qt, kt, vt, ot);
  mlstm_gates<<<Bsz, 256, 0, stream>>>(x, wi, bi, wf, bf, itp, ftp);
  mlstm_nstate<<<Bsz, 256, 0, stream>>>(n_prev, kt, qt, itp, ftp, n_out, invp);
  mlstm_cstream<<<Bsz * (Msz / 8), 256, 0, stream>>>(C_prev, qt, kt, vt, ot,
                                                     itp, ftp, invp, C, ht);
}